// VolumeRenderer_40398462386540
// MI455X (gfx1250) — compile-verified
//
#include <hip/hip_runtime.h>
#include <stdint.h>

// Volume rendering (NeRF-style) for MI455X / gfx1250.
// Memory-bound: ~336 MB read per call -> ~14.5 us floor at 23.3 TB/s.
// Design: 1 ray per wave32, shuffle-based exclusive cumprod scan,
// feature rows staged global->LDS with CDNA5 async-tensor path
// (global_load_async_to_lds_b128 + s_wait_asynccnt) overlapped with compute.

#define NPTS      128
#define FAR_DELTA 1e10f
#define WPB       8            // waves (rays) per block
#define TPB       (WPB * 32)   // 256 threads

__global__ __launch_bounds__(TPB) void volrender_kernel(
    const float* __restrict__ depth_values,  // [n, 128]
    const float* __restrict__ density,       // [n, 128]
    const float* __restrict__ feature,       // [n, 128, 3]
    float* __restrict__ out,                 // [n*3] feat, then [n] depth
    int n_rays)
{
    __shared__ __align__(16) float sfeat[WPB * NPTS * 3];  // 12 KB: 1536 B per wave

    const int lane = threadIdx.x & 31;
    const int wv   = threadIdx.x >> 5;
    const int ray  = blockIdx.x * WPB + wv;
    if (ray >= n_rays) return;   // uniform per wave

    // ---------------- async stage: this ray's feature row (1536 B) -> LDS ----
    // Three wave-level async b128 transfers, each 512 B unit-stride coalesced.
    // INST_OFFSET is added to both the global and the LDS address (ISA 8.4.4),
    // so one base register pair covers all three.
    const char* gfeat = (const char*)feature
                      + (size_t)ray * (NPTS * 3 * sizeof(float)) + (size_t)(lane * 16);
    float* sbase = &sfeat[wv * (NPTS * 3)];
    uint32_t lds_addr = (uint32_t)(uintptr_t)sbase + (uint32_t)(lane * 16);
    asm volatile(
        "global_load_async_to_lds_b128 %0, %1, off\n\t"
        "global_load_async_to_lds_b128 %0, %1, off offset:512\n\t"
        "global_load_async_to_lds_b128 %0, %1, off offset:1024"
        :: "v"(lds_addr), "v"(gfeat)
        : "memory");

    // ---------------- direct loads: depth + density (coalesced b128) ---------
    const float4* dptr = (const float4*)(depth_values + (size_t)ray * NPTS) + lane;
    const float4* gptr = (const float4*)(density      + (size_t)ray * NPTS) + lane;
    float4 d4 = *dptr;   // samples 4L..4L+3
    float4 g4 = *gptr;

    // deltas; last sample of the ray gets FAR_DELTA
    float dnext = __shfl_down(d4.x, 1);  // depth[4L+4] from neighbor lane
    float del0 = d4.y - d4.x;
    float del1 = d4.z - d4.y;
    float del2 = d4.w - d4.z;
    float del3 = (lane == 31) ? FAR_DELTA : (dnext - d4.w);

    // t_i = exp(-relu(sigma)*delta) + 1e-10 ;  alpha_i = 1 - exp(...)
    float e0 = __expf(-fmaxf(g4.x, 0.0f) * del0);
    float e1 = __expf(-fmaxf(g4.y, 0.0f) * del1);
    float e2 = __expf(-fmaxf(g4.z, 0.0f) * del2);
    float e3 = __expf(-fmaxf(g4.w, 0.0f) * del3);
    float a0 = 1.0f - e0, a1 = 1.0f - e1, a2 = 1.0f - e2, a3 = 1.0f - e3;
    float t0 = e0 + 1e-10f, t1 = e1 + 1e-10f, t2 = e2 + 1e-10f, t3 = e3 + 1e-10f;

    // per-lane exclusive prefix products
    float p1 = t0;
    float p2 = p1 * t1;
    float p3 = p2 * t2;
    float q  = p3 * t3;   // product of this lane's 4 factors

    // wave-wide multiplicative inclusive scan of q (Hillis-Steele, 5 steps)
    float v = q;
#pragma unroll
    for (int off = 1; off < 32; off <<= 1) {
        float nv = __shfl_up(v, off);
        if (lane >= off) v *= nv;
    }
    // exclusive: product of all t_j with j < 4L
    float Tb = __shfl_up(v, 1);
    if (lane == 0) Tb = 1.0f;

    float w0 = a0 * Tb;
    float w1 = a1 * (Tb * p1);
    float w2 = a2 * (Tb * p2);
    float w3 = a3 * (Tb * p3);

    float dep = w0 * d4.x + w1 * d4.y + w2 * d4.z + w3 * d4.w;

    // ---------------- consume staged features from LDS -----------------------
    asm volatile("s_wait_asynccnt 0" ::: "memory");
    const float4* sf = (const float4*)(sbase + lane * 12);  // 48B-aligned
    float4 f0 = sf[0];  // s0.r s0.g s0.b s1.r
    float4 f1 = sf[1];  // s1.g s1.b s2.r s2.g
    float4 f2 = sf[2];  // s2.b s3.r s3.g s3.b
    float fr = w0 * f0.x + w1 * f0.w + w2 * f1.z + w3 * f2.y;
    float fg = w0 * f0.y + w1 * f1.x + w2 * f1.w + w3 * f2.z;
    float fb = w0 * f0.z + w1 * f1.y + w2 * f2.x + w3 * f2.w;

    // ---------------- wave reduction of (fr, fg, fb, dep) --------------------
#pragma unroll
    for (int off = 16; off >= 1; off >>= 1) {
        fr  += __shfl_down(fr,  off);
        fg  += __shfl_down(fg,  off);
        fb  += __shfl_down(fb,  off);
        dep += __shfl_down(dep, off);
    }

    if (lane == 0) {
        float* of = out + (size_t)ray * 3;
        of[0] = fr;
        of[1] = fg;
        of[2] = fb;
        out[(size_t)n_rays * 3 + ray] = dep;
    }
}

extern "C" void kernel_launch(void* const* d_in, const int* in_sizes, int n_in,
                              void* d_out, int out_size, void* d_ws, size_t ws_size,
                              hipStream_t stream) {
    const float* depth = (const float*)d_in[0];   // [N,128] f32
    const float* dens  = (const float*)d_in[1];   // [N,128] f32
    const float* feat  = (const float*)d_in[2];   // [N,128,3] f32
    float* out = (float*)d_out;                   // [N*3] + [N]

    int n_rays = in_sizes[0] / NPTS;
    int blocks = (n_rays + WPB - 1) / WPB;
    volrender_kernel<<<blocks, TPB, 0, stream>>>(depth, dens, feat, out, n_rays);
}